// GraphSAGELayer_11038065951060
// MI455X (gfx1250) — compile-verified
//
#include <hip/hip_runtime.h>

typedef __attribute__((ext_vector_type(2))) float v2f;
typedef __attribute__((ext_vector_type(8))) float v8f;

#define FD 128          // feature dim D
#define FD2 256         // 2*D
#define LDS_STRIDE 258  // 256 + 2 pad: breaks 16-way LDS bank conflict on A-frag reads

// ---------------------------------------------------------------------------
// Kernel 1: zero the workspace (neigh_sum + deg)
// ---------------------------------------------------------------------------
__global__ void zero_ws_kernel(float* __restrict__ ws, long n) {
    long i = (long)blockIdx.x * blockDim.x + threadIdx.x;
    long stride = (long)gridDim.x * blockDim.x;
    for (; i < n; i += stride) ws[i] = 0.0f;
}

// ---------------------------------------------------------------------------
// Kernel 2: edge scatter. One wave32 per edge: lane l covers feature cols
// 4l..4l+3 (float4 gather, 4 hardware f32 atomic adds). Lane 0 bumps degree.
// x fits in L2 (25.6MB << 192MB) so gathers + atomics stay on-chip.
// ---------------------------------------------------------------------------
__global__ __launch_bounds__(256) void scatter_kernel(
    const float* __restrict__ x,
    const int* __restrict__ src,
    const int* __restrict__ dst,
    float* __restrict__ nsum,
    float* __restrict__ deg,
    int n_edges) {
    int lane = threadIdx.x & 31;
    int edge = blockIdx.x * (blockDim.x >> 5) + (threadIdx.x >> 5);
    if (edge >= n_edges) return;
    int s = src[edge];
    int d = dst[edge];
    const float4* xs = (const float4*)(x + (long)s * FD);
    float4 v = xs[lane];  // coalesced 512B per wave
    float* base = nsum + (long)d * FD + 4 * lane;
    unsafeAtomicAdd(base + 0, v.x);
    unsafeAtomicAdd(base + 1, v.y);
    unsafeAtomicAdd(base + 2, v.z);
    unsafeAtomicAdd(base + 3, v.w);
    if (lane == 0) unsafeAtomicAdd(deg + d, 1.0f);
}

// ---------------------------------------------------------------------------
// Kernel 3: out = relu([x | agg] @ W^T + b) using V_WMMA_F32_16X16X4_F32.
// Block = 256 threads (8 waves). Block owns 16 output rows (M-tile); the
// 16x256 h-tile (x concat mean-aggregate with deg==0 fallback) is staged in
// LDS once. Wave w computes output cols [16w, 16w+16); K-loop of 64 WMMAs.
// Bias is folded into the WMMA C-initializer (C/D rows share lane-indexed N).
// ---------------------------------------------------------------------------
__global__ __launch_bounds__(256) void sage_gemm_kernel(
    const float* __restrict__ x,
    const float* __restrict__ W,      // (128, 256) row-major
    const float* __restrict__ bias,   // (128,)
    const float* __restrict__ nsum,
    const float* __restrict__ deg,
    float* __restrict__ out,          // (N, 128)
    int n_nodes) {
    __shared__ float h[16 * LDS_STRIDE];

    const int tid = threadIdx.x;
    const int m0 = blockIdx.x * 16;

    // Cooperative, coalesced load of h = [x | agg] for 16 rows.
    for (int idx = tid; idx < 16 * FD2; idx += 256) {
        int r = idx >> 8;     // 0..15
        int c = idx & 255;    // 0..255
        int node = m0 + r;
        float v = 0.0f;
        if (node < n_nodes) {
            if (c < FD) {
                v = x[(long)node * FD + c];
            } else {
                int cc = c - FD;
                float dg = deg[node];
                float sm = nsum[(long)node * FD + cc];
                float xv = x[(long)node * FD + cc];
                v = (dg > 0.0f) ? (sm / dg) : xv;   // mean, or self if isolated
            }
        }
        h[r * LDS_STRIDE + c] = v;
    }
    __syncthreads();

    const int lane = tid & 31;
    const int wv   = tid >> 5;        // 0..7  -> N tile
    const int half = lane >> 4;       // 0/1   -> K-pair select (ISA A/B layout)
    const int lmod = lane & 15;       // A: M row; B/D: N col
    const int j    = wv * 16 + lmod;  // output column

    const float* wrow = W + (long)j * FD2 + 2 * half;         // W[j, k..k+1]
    const float* arow = &h[lmod * LDS_STRIDE + 2 * half];     // h[m, k..k+1]

    // Fold bias into the accumulator: every M-row of column j gets bias[j].
    float bj = bias[j];
    v8f acc = {bj, bj, bj, bj, bj, bj, bj, bj};

#pragma unroll 8
    for (int k0 = 0; k0 < FD2; k0 += 4) {
        v2f a = *(const v2f*)(arow + k0);   // ds_load_2addr_b64, conflict-free
        v2f b = *(const v2f*)(wrow + k0);   // global_load_b64, L2-resident
        acc = __builtin_amdgcn_wmma_f32_16x16x4_f32(
            /*neg_a=*/false, a, /*neg_b=*/false, b,
            /*c_mod=*/(short)0, acc, /*reuse_a=*/false, /*reuse_b=*/false);
    }

    // D layout: VGPR i -> M = i + 8*half, N = lmod. ReLU + store.
    const int mbase = m0 + 8 * half;
    if (m0 + 16 <= n_nodes) {
        // Full tile (all blocks except possibly the last): branch-free stores.
#pragma unroll
        for (int i = 0; i < 8; ++i) {
            float val = acc[i];
            out[(long)(mbase + i) * FD + j] = val > 0.0f ? val : 0.0f;
        }
    } else {
        // Partial last tile: guarded scalar stores.
#pragma unroll
        for (int i = 0; i < 8; ++i) {
            int m = mbase + i;
            if (m < n_nodes) {
                float val = acc[i];
                out[(long)m * FD + j] = val > 0.0f ? val : 0.0f;
            }
        }
    }
}

// ---------------------------------------------------------------------------
extern "C" void kernel_launch(void* const* d_in, const int* in_sizes, int n_in,
                              void* d_out, int out_size, void* d_ws, size_t ws_size,
                              hipStream_t stream) {
    const float* x    = (const float*)d_in[0];  // (N, 128)
    const float* W    = (const float*)d_in[1];  // (128, 256)
    const float* bias = (const float*)d_in[2];  // (128,)
    const int*   src  = (const int*)d_in[3];    // (E,)
    const int*   dst  = (const int*)d_in[4];    // (E,)

    const int n_nodes = in_sizes[0] / FD;
    const int n_edges = in_sizes[3];

    float* nsum = (float*)d_ws;                       // N*128 floats
    float* deg  = nsum + (long)n_nodes * FD;          // N floats
    float* outf = (float*)d_out;

    long zn = (long)n_nodes * FD + n_nodes;
    zero_ws_kernel<<<1024, 256, 0, stream>>>(nsum, zn);

    const int edges_per_block = 256 / 32;
    scatter_kernel<<<(n_edges + edges_per_block - 1) / edges_per_block, 256, 0, stream>>>(
        x, src, dst, nsum, deg, n_edges);

    sage_gemm_kernel<<<(n_nodes + 15) / 16, 256, 0, stream>>>(
        x, W, bias, nsum, deg, outf, n_nodes);
}